// ReachabilityModel_83408264888623
// MI455X (gfx1250) — compile-verified
//
#include <hip/hip_runtime.h>
#include <hip/hip_bf16.h>

// ============================================================================
// ReachabilityModel: out[t] = sum_k coeffs[t,k] * A^k @ onehot(batch)
//   N_NODES=100000, N_EDGES=3.2M, BATCH=256, NUM_TAU=4, ORDER=8
//
// Strategy (MI455X, compile-only reasoning):
//  * SpMM is memory bound; monome (102.4MB) is L2-resident (192MB L2).
//    CSR-by-dst built on device -> register accumulation, no fp atomics.
//  * Coefficient combine = (4 x 4chunk) @ (4chunk x 25.6M) GEMM, executed on
//    the matrix pipe with V_WMMA_F32_16X16X4_F32 (f32 accumulate, exact
//    precision match). Two passes (k=1..4, k=5..8) => out RMW'd only twice.
// ============================================================================

#define NUM_TAU 4
#define NCOEF   9   // ORDER + 1
#define BCOLS   256

typedef __attribute__((ext_vector_type(2))) float v2f;
typedef __attribute__((ext_vector_type(8))) float v8f;

// ---------------------------------------------------------------- CSR build
__global__ void count_kernel(const int* __restrict__ dst, int* __restrict__ counts,
                             int n_edges) {
    int e = blockIdx.x * blockDim.x + threadIdx.x;
    if (e < n_edges) atomicAdd(&counts[dst[e]], 1);
}

// single workgroup of 1024 threads: exclusive scan of counts -> row_ptr
__global__ void scan_kernel(const int* __restrict__ counts, int* __restrict__ row_ptr,
                            int n_nodes) {
    __shared__ int partial[1024];
    const int tid   = threadIdx.x;
    const int chunk = (n_nodes + 1023) / 1024;
    const int begin = tid * chunk;
    const int end   = min(begin + chunk, n_nodes);
    int s = 0;
    for (int i = begin; i < end; ++i) s += counts[i];
    partial[tid] = s;
    __syncthreads();
    for (int off = 1; off < 1024; off <<= 1) {
        int add = (tid >= off) ? partial[tid - off] : 0;
        __syncthreads();
        partial[tid] += add;
        __syncthreads();
    }
    int run = (tid == 0) ? 0 : partial[tid - 1];
    for (int i = begin; i < end; ++i) { row_ptr[i] = run; run += counts[i]; }
    if (tid == 1023) row_ptr[n_nodes] = partial[1023];
}

__global__ void fill_kernel(const int* __restrict__ dst, const int* __restrict__ src,
                            const float* __restrict__ ew, const int* __restrict__ row_ptr,
                            int* __restrict__ cursor, float* __restrict__ ew_s,
                            int* __restrict__ src_s, int n_edges) {
    int e = blockIdx.x * blockDim.x + threadIdx.x;
    if (e >= n_edges) return;
    int d = dst[e];
    int p = row_ptr[d] + atomicAdd(&cursor[d], 1);
    ew_s[p]  = ew[e];
    src_s[p] = src[e];
}

// ------------------------------------------- out init + one-hot M0 scatter
__global__ void init_scatter(const int* __restrict__ bidx,
                             const float* __restrict__ coeffs,
                             float* __restrict__ out, float* __restrict__ M0,
                             int n_nodes, int batch) {
    int j = blockIdx.x * blockDim.x + threadIdx.x;
    if (j >= batch) return;
    int node = bidx[j];
    size_t plane = (size_t)n_nodes * BCOLS;
    size_t off   = (size_t)node * BCOLS + j;
    M0[off] = 1.0f;
    for (int t = 0; t < NUM_TAU; ++t) out[t * plane + off] = coeffs[t * NCOEF + 0];
}

// --------------------------------------------------------- CSR SpMM (wave/row)
// wave32: lane owns columns [lane*8, lane*8+8); (w,src) staged via LDS so the
// inner loop is broadcast ds_load + two coalesced b128 gathers per edge.
__global__ void __launch_bounds__(256)
spmm_csr(const float4* __restrict__ Min, float4* __restrict__ Mout,
         const int* __restrict__ row_ptr, const float* __restrict__ ew_s,
         const int* __restrict__ src_s, int n_nodes) {
    __shared__ float sw[8 * 32];
    __shared__ int   ss[8 * 32];
    const int wid  = threadIdx.x >> 5;
    const int lane = threadIdx.x & 31;
    const int row  = blockIdx.x * 8 + wid;
    if (row >= n_nodes) return;
    const int b0 = row_ptr[row], b1 = row_ptr[row + 1];
    float4 a0 = {0.f, 0.f, 0.f, 0.f}, a1 = {0.f, 0.f, 0.f, 0.f};
    const int base = wid * 32;
    for (int cs = b0; cs < b1; cs += 32) {
        int e = cs + lane;
        float w = 0.f; int s = 0;
        if (e < b1) { w = ew_s[e]; s = src_s[e]; }
        sw[base + lane] = w;
        ss[base + lane] = s;
        __builtin_amdgcn_wave_barrier();   // per-wave LDS ordering (in-order DS)
        const int cnt = min(32, b1 - cs);
        for (int i = 0; i < cnt; ++i) {
            const float wi = sw[base + i];
            const int   si = ss[base + i];
            const float4* rp = Min + (size_t)si * 64 + lane * 2;
            const float4 r0 = rp[0], r1 = rp[1];
            a0.x = fmaf(wi, r0.x, a0.x); a0.y = fmaf(wi, r0.y, a0.y);
            a0.z = fmaf(wi, r0.z, a0.z); a0.w = fmaf(wi, r0.w, a0.w);
            a1.x = fmaf(wi, r1.x, a1.x); a1.y = fmaf(wi, r1.y, a1.y);
            a1.z = fmaf(wi, r1.z, a1.z); a1.w = fmaf(wi, r1.w, a1.w);
        }
        __builtin_amdgcn_wave_barrier();
    }
    float4* op = Mout + (size_t)row * 64 + lane * 2;
    op[0] = a0; op[1] = a1;
}

// ----------------------------------------------- WMMA combine (4 terms/pass)
// out[t, n, c] += sum_{k=0..3} coeffs[t, k0+k] * Mk[n, c]
// Mapped onto V_WMMA_F32_16X16X4_F32: M=tau(pad 16), K=4 terms, N=16 cols.
// A 16x4 layout (ISA 7.12.2): lanes 0-15 -> {K0,K1}, lanes 16-31 -> {K2,K3};
// B 4x16 mirrored; C/D 16x16: VGPR r holds rows r (lanes 0-15) / r+8 (16-31).
__global__ void __launch_bounds__(256)
combine_wmma(const float* __restrict__ coeffs,
             const float* __restrict__ M0, const float* __restrict__ M1,
             const float* __restrict__ M2, const float* __restrict__ M3,
             float* __restrict__ out, int n_nodes, int k0) {
    const int wave = (blockIdx.x * blockDim.x + threadIdx.x) >> 5;
    const int lane = threadIdx.x & 31;
    if (wave >= n_nodes) return;
    const int  half = lane >> 4;     // 0: K0,K1   1: K2,K3
    const int  m    = lane & 15;
    v2f a = {0.f, 0.f};
    if (m < NUM_TAU) {
        a.x = coeffs[m * NCOEF + k0 + half * 2 + 0];
        a.y = coeffs[m * NCOEF + k0 + half * 2 + 1];
    }
    const float* Bk0 = half ? M2 : M0;
    const float* Bk1 = half ? M3 : M1;
    const size_t rowOff = (size_t)wave * BCOLS;
    const size_t plane  = (size_t)n_nodes * BCOLS;
    for (int cb = 0; cb < 16; ++cb) {
        const size_t col = rowOff + cb * 16 + m;
        v2f b;
        b.x = Bk0[col];
        b.y = Bk1[col];
        v8f c = {0.f, 0.f, 0.f, 0.f, 0.f, 0.f, 0.f, 0.f};
        if (lane < 16) {
            c[0] = out[0 * plane + col];
            c[1] = out[1 * plane + col];
            c[2] = out[2 * plane + col];
            c[3] = out[3 * plane + col];
        }
        v8f d = __builtin_amdgcn_wmma_f32_16x16x4_f32(
            /*neg_a=*/false, a, /*neg_b=*/false, b,
            /*c_mod=*/(short)0, c, /*reuse_a=*/false, /*reuse_b=*/false);
        if (lane < 16) {
            out[0 * plane + col] = d[0];
            out[1 * plane + col] = d[1];
            out[2 * plane + col] = d[2];
            out[3 * plane + col] = d[3];
        }
    }
}

// ============================================================================
extern "C" void kernel_launch(void* const* d_in, const int* in_sizes, int n_in,
                              void* d_out, int out_size, void* d_ws, size_t ws_size,
                              hipStream_t stream) {
    const float* coeffs = (const float*)d_in[0];
    const float* ew     = (const float*)d_in[1];
    const int*   src    = (const int*)d_in[2];
    const int*   dst    = (const int*)d_in[3];
    const int*   bidx   = (const int*)d_in[4];
    const int n_edges   = in_sizes[1];
    const int batch     = in_sizes[4];                 // 256
    const int n_nodes   = out_size / (NUM_TAU * BCOLS);
    float* out = (float*)d_out;

    // ---- workspace carve-up (5 ping-pong slots + CSR arrays) ----
    const size_t slotE = (size_t)n_nodes * BCOLS;      // 25.6M floats
    float* slots  = (float*)d_ws;
    float* slot0  = slots + 0 * slotE;
    float* slot1  = slots + 1 * slotE;
    float* slot2  = slots + 2 * slotE;
    float* slot3  = slots + 3 * slotE;
    float* slot4  = slots + 4 * slotE;                 // M0 one-hot, later M5
    float* ew_s   = slots + 5 * slotE;
    int*   src_s  = (int*)(ew_s + n_edges);
    int*   rowp   = src_s + n_edges;
    int*   counts = rowp + (n_nodes + 1);

    const int TB = 256;
    const int edgeBlocks = (n_edges + TB - 1) / TB;
    const int rowBlocks  = (n_nodes + 7) / 8;          // 8 waves (rows) per block

    // ---- init ----
    hipMemsetAsync(d_out, 0, (size_t)out_size * sizeof(float), stream);
    hipMemsetAsync(slot4, 0, slotE * sizeof(float), stream);
    hipMemsetAsync(counts, 0, (size_t)n_nodes * sizeof(int), stream);

    // ---- CSR by dst ----
    count_kernel<<<edgeBlocks, TB, 0, stream>>>(dst, counts, n_edges);
    scan_kernel<<<1, 1024, 0, stream>>>(counts, rowp, n_nodes);
    hipMemsetAsync(counts, 0, (size_t)n_nodes * sizeof(int), stream); // cursors
    fill_kernel<<<edgeBlocks, TB, 0, stream>>>(dst, src, ew, rowp, counts,
                                               ew_s, src_s, n_edges);

    // ---- out = coeffs[:,0] * onehot ; M0 = onehot ----
    init_scatter<<<(batch + TB - 1) / TB, TB, 0, stream>>>(bidx, coeffs, out,
                                                           slot4, n_nodes, batch);

    // ---- k = 1..4 : SpMM chain, then WMMA combine pass 1 ----
    spmm_csr<<<rowBlocks, TB, 0, stream>>>((const float4*)slot4, (float4*)slot0,
                                           rowp, ew_s, src_s, n_nodes); // M1
    spmm_csr<<<rowBlocks, TB, 0, stream>>>((const float4*)slot0, (float4*)slot1,
                                           rowp, ew_s, src_s, n_nodes); // M2
    spmm_csr<<<rowBlocks, TB, 0, stream>>>((const float4*)slot1, (float4*)slot2,
                                           rowp, ew_s, src_s, n_nodes); // M3
    spmm_csr<<<rowBlocks, TB, 0, stream>>>((const float4*)slot2, (float4*)slot3,
                                           rowp, ew_s, src_s, n_nodes); // M4
    combine_wmma<<<rowBlocks, TB, 0, stream>>>(coeffs, slot0, slot1, slot2, slot3,
                                               out, n_nodes, /*k0=*/1);

    // ---- k = 5..8 : SpMM chain (reusing freed slots), combine pass 2 ----
    spmm_csr<<<rowBlocks, TB, 0, stream>>>((const float4*)slot3, (float4*)slot4,
                                           rowp, ew_s, src_s, n_nodes); // M5
    spmm_csr<<<rowBlocks, TB, 0, stream>>>((const float4*)slot4, (float4*)slot0,
                                           rowp, ew_s, src_s, n_nodes); // M6
    spmm_csr<<<rowBlocks, TB, 0, stream>>>((const float4*)slot0, (float4*)slot1,
                                           rowp, ew_s, src_s, n_nodes); // M7
    spmm_csr<<<rowBlocks, TB, 0, stream>>>((const float4*)slot1, (float4*)slot2,
                                           rowp, ew_s, src_s, n_nodes); // M8
    combine_wmma<<<rowBlocks, TB, 0, stream>>>(coeffs, slot4, slot0, slot1, slot2,
                                               out, n_nodes, /*k0=*/5);
}